// GaussianMixture_19215683682749
// MI455X (gfx1250) — compile-verified
//
#include <hip/hip_runtime.h>
#include <hip/hip_bf16.h>
#include <math.h>

typedef float v2f __attribute__((ext_vector_type(2)));
typedef float v8f __attribute__((ext_vector_type(8)));

#define DIM   20
#define KC    25
#define KPAD  32

// workspace layout (floats)
#define WS_PP    0                      // [-0.5*exp(nlv)]  (KPAD x DIM)
#define WS_Q     (KPAD * DIM)           // [mean*exp(nlv)]  (KPAD x DIM)
#define WS_C     (2 * KPAD * DIM)       // per-comp constant (KPAD)
#define WS_PRIOR (2 * KPAD * DIM + KPAD)

#define NLV0   1.3862943611198906f      // -2*log(0.5)
#define LOG2PI 1.8378770664093453f

// ---------------------------------------------------------------------------
// Setup: fold everything component-wise into GEMM operands + scalars.
// ---------------------------------------------------------------------------
__global__ void gmm_setup(const float* __restrict__ mean,
                          const float* __restrict__ nlv,
                          const float* __restrict__ w,
                          float* __restrict__ ws) {
    __shared__ float s_m2p[KPAD];   // sum mu^2 * p
    __shared__ float s_nlv[KPAD];   // sum nlv
    __shared__ float s_mn2[KPAD];   // ||mu||^2
    __shared__ float s_nvlp[KPAD];  // neglogvar prior partial

    const int k = threadIdx.x;      // 32 threads
    float m2p = 0.f, nlvs = 0.f, mn2 = 0.f, nvlp = 0.f;
    if (k < KC) {
        for (int d = 0; d < DIM; ++d) {
            float nv = nlv[k * DIM + d];
            float mu = mean[k * DIM + d];
            float p  = expf(nv);
            ws[WS_PP + k * DIM + d] = -0.5f * p;
            ws[WS_Q  + k * DIM + d] = mu * p;
            m2p  += mu * mu * p;
            nlvs += nv;
            mn2  += mu * mu;
            float dn = nv - NLV0;                 // SD_SD == 1
            nvlp += -0.5f * LOG2PI - 0.5f * dn * dn;
        }
    } else {
        for (int d = 0; d < DIM; ++d) {           // zero padding comps
            ws[WS_PP + k * DIM + d] = 0.f;
            ws[WS_Q  + k * DIM + d] = 0.f;
        }
    }
    s_m2p[k] = m2p; s_nlv[k] = nlvs; s_mn2[k] = mn2; s_nvlp[k] = nvlp;
    __syncthreads();

    if (k == 0) {
        const float PI_TERM       = -0.5f * DIM * LOG2PI;
        const float DIRICHLET     = lgammaf((float)KC);
        const float SOFTBALL_NORM = lgammaf(1.0f + DIM * 0.5f)
                                  - DIM * (logf(2.0f) + 0.5f * logf((float)M_PI));
        // log-softmax of weights
        float wm = -3.4e38f;
        for (int i = 0; i < KC; ++i) wm = fmaxf(wm, w[i]);
        float sw = 0.f;
        for (int i = 0; i < KC; ++i) sw += expf(w[i] - wm);
        float lsew = wm + logf(sw);

        float prior = DIRICHLET;
        for (int i = 0; i < KC; ++i) {
            ws[WS_C + i] = -0.5f * s_m2p[i] + 0.5f * s_nlv[i] + PI_TERM + (w[i] - lsew);
            float z = 5.0f * (sqrtf(s_mn2[i]) * 0.5f - 1.0f);   // SHARP*(|mu|/R - 1)
            float softplus = (z > 20.f) ? z : log1pf(expf(z));
            prior += SOFTBALL_NORM - softplus + s_nvlp[i];
        }
        for (int i = KC; i < KPAD; ++i) ws[WS_C + i] = 0.f;
        ws[WS_PRIOR] = prior;
    }
}

// ---------------------------------------------------------------------------
// Main: per wave, one 16-row tile. y = X2*Pp + X*Q + C via WMMA f32 16x16x4,
// then row-wise logsumexp through LDS.
// ---------------------------------------------------------------------------
__global__ __launch_bounds__(256) void gmm_main(const float* __restrict__ x,
                                                const float* __restrict__ ws,
                                                float* __restrict__ out,
                                                int N) {
    __shared__ float smem[8][16][33];   // per-wave 16x32 result, padded stride

    const int lane = threadIdx.x & 31;
    const int wv   = threadIdx.x >> 5;
    const int mrow = lane & 15;         // M (A) or N (B) within tile
    const int hi   = lane >> 4;         // 0: K=0,1 ; 1: K=2,3 within a k-step
    const int dofs = hi * 2;
    const int row0 = (blockIdx.x * 8 + wv) * 16;

    // B operands: 5 k-steps x 2 column tiles, float2 per lane (K pair)
    v2f bp[5][2], bq[5][2];
#pragma unroll
    for (int s = 0; s < 5; ++s) {
#pragma unroll
        for (int t = 0; t < 2; ++t) {
            int idx = (t * 16 + mrow) * DIM + 4 * s + dofs;
            bp[s][t] = *(const v2f*)(ws + WS_PP + idx);
            bq[s][t] = *(const v2f*)(ws + WS_Q  + idx);
        }
    }

    int row = row0 + mrow;
    if (row >= N) row = N - 1;          // clamp: keep EXEC all-ones for WMMA
    const float* xr = x + (size_t)row * DIM + dofs;

    v8f acc0 = {};
    v8f acc1 = {};
#pragma unroll
    for (int s = 0; s < 5; ++s) {
        v2f a  = *(const v2f*)(xr + 4 * s);
        v2f a2 = a * a;
        acc0 = __builtin_amdgcn_wmma_f32_16x16x4_f32(false, a2, false, bp[s][0],
                                                     (short)0, acc0, false, false);
        acc0 = __builtin_amdgcn_wmma_f32_16x16x4_f32(false, a,  false, bq[s][0],
                                                     (short)0, acc0, false, false);
        acc1 = __builtin_amdgcn_wmma_f32_16x16x4_f32(false, a2, false, bp[s][1],
                                                     (short)0, acc1, false, false);
        acc1 = __builtin_amdgcn_wmma_f32_16x16x4_f32(false, a,  false, bq[s][1],
                                                     (short)0, acc1, false, false);
    }

    // add per-component constant, stash to LDS (C/D layout: VGPR i -> M = i + 8*hi)
    float c0 = ws[WS_C + mrow];
    float c1 = ws[WS_C + 16 + mrow];
#pragma unroll
    for (int i = 0; i < 8; ++i) {
        smem[wv][i + hi * 8][mrow]      = acc0[i] + c0;
        smem[wv][i + hi * 8][16 + mrow] = acc1[i] + c1;
    }
    __syncthreads();

    const float prior = ws[WS_PRIOR];
    if (lane < 16) {
        float mx = -3.4e38f;
#pragma unroll
        for (int k = 0; k < KC; ++k) mx = fmaxf(mx, smem[wv][lane][k]);
        float sum = 0.f;
#pragma unroll
        for (int k = 0; k < KC; ++k) sum += __expf(smem[wv][lane][k] - mx);
        int r = row0 + lane;
        if (r < N) out[r] = -(mx + __logf(sum) + prior);
    }
}

// ---------------------------------------------------------------------------
extern "C" void kernel_launch(void* const* d_in, const int* in_sizes, int n_in,
                              void* d_out, int out_size, void* d_ws, size_t ws_size,
                              hipStream_t stream) {
    const float* x    = (const float*)d_in[0];
    const float* mean = (const float*)d_in[1];
    const float* nlv  = (const float*)d_in[2];
    const float* w    = (const float*)d_in[3];
    float* out = (float*)d_out;
    float* ws  = (float*)d_ws;

    const int N = in_sizes[0] / DIM;

    gmm_setup<<<1, 32, 0, stream>>>(mean, nlv, w, ws);
    const int blocks = (N + 127) / 128;   // 128 rows per block (8 waves x 16)
    gmm_main<<<blocks, 256, 0, stream>>>(x, ws, out, N);
}